// WideDeep_47880295416088
// MI455X (gfx1250) — compile-verified
//
#include <hip/hip_runtime.h>
#include <hip/hip_bf16.h>

typedef __attribute__((ext_vector_type(2))) float v2f;
typedef __attribute__((ext_vector_type(8))) float v8f;

#define LR          0.05f
// log2(1 - 1e-6): lazy-decay exponent base in log2 space
#define LOG2_DECAY  (-1.4426952216940865e-06f)
#define LOG2E       (1.4426950408889634f)
#define HSIZE       16384   // 64 KB LDS hash (WGP has 320 KB); dup/collision detector

// One wave32 runs the whole sequential scan. Per sample:
//   lanes hold 2 indices each (64 total, one b64 load) -> gather w/dt -> decay ->
//   WMMA f32 16x16x4 vs ones() collapses the 64-way f32 sum ->
//   fast sigmoid/grad (raw v_exp + v_rcp) ->
//   LDS-hash uniqueness fast path (exact; falls back to shuffle dedup) ->
//   scatter update, fence, next sample.
__global__ __launch_bounds__(32, 1)
void wide_sgd_seq_kernel(const int* __restrict__ idx,
                         const float* __restrict__ y,
                         const float* __restrict__ b_in,
                         float* __restrict__ w,
                         int* __restrict__ dt,
                         float* __restrict__ out,
                         int B)
{
    __shared__ int htab[HSIZE];

    const int lane = threadIdx.x;          // 0..31, wave32
    float b = b_in[0];

    v2f ones; ones.x = 1.0f; ones.y = 1.0f; // B-matrix: all ones (4x16)

    for (int j = 0; j < B; ++j) {
        // one global_load_b64 fetches both of this lane's indices
        const int2 ii = *(const int2*)(idx + j * 64 + 2 * lane);
        const int i0 = ii.x;
        const int i1 = ii.y;

        // gather weights + timestamps (L2-resident; 4 independent loads in flight)
        const float w0v = w[i0];
        const float w1v = w[i1];
        const int   d0  = dt[i0];
        const int   d1  = dt[i1];

        // lazy L2 decay: (1-lambda)^(q - dt[i]), q == j.
        // |arg| <= 1.5e-3 -> raw v_exp_f32 is exact enough, no range fixup.
        const float wv0 = w0v * __builtin_amdgcn_exp2f((float)(j - d0) * LOG2_DECAY);
        const float wv1 = w1v * __builtin_amdgcn_exp2f((float)(j - d1) * LOG2_DECAY);

        // ---- 64-wide f32 reduction on the matrix unit ----
        // A = 16x4 f32 (2 VGPRs/lane = our 64 decayed values), B = ones(4x16)
        // D[m][n] = rowsum(A[m]); lanes 0-15 carry M=0..7, lanes 16-31 carry M=8..15
        v2f A; A.x = wv0; A.y = wv1;
        v8f acc = {};
        acc = __builtin_amdgcn_wmma_f32_16x16x4_f32(false, A, false, ones,
                                                    (short)0, acc, false, false);
        const float s01 = acc[0] + acc[1], s23 = acc[2] + acc[3];
        const float s45 = acc[4] + acc[5], s67 = acc[6] + acc[7];
        const float rs  = (s01 + s23) + (s45 + s67);    // half-wave partial sum
        const float total = rs + __shfl_xor(rs, 16);    // full 64-value sum, all lanes

        // sigmoid + gradient, redundantly on every lane (keeps EXEC uniform).
        // |z| <= 35 -> |z*log2e| <= 50.5: raw exp2 is safe; rcp is ~1 ulp.
        const float z  = fminf(fmaxf(total + b, -35.0f), 35.0f);
        const float e  = __builtin_amdgcn_exp2f(-z * LOG2E);
        const float p  = __builtin_amdgcn_rcpf(1.0f + e);
        const float g  = p - y[j];
        const float lg = LR * g;

        if (lane == 0) out[j] = p;

        // ---- exact uniqueness fast path via LDS hash ----
        // Tag is unique per (sample, slot). If every slot reads back its own tag,
        // all 64 buckets are distinct => all 64 indices are distinct.
        const int h0   = i0 & (HSIZE - 1);
        const int h1   = i1 & (HSIZE - 1);
        const int tag0 = (j << 7) | (2 * lane);
        const int tag1 = (j << 7) | (2 * lane + 1);
        htab[h0] = tag0;
        htab[h1] = tag1;
        __syncthreads();                    // single wave: cheap; orders DS ops
        // bitwise & (not &&): both ds_loads issue together, no divergent branch
        const bool uniq = (htab[h0] == tag0) & (htab[h1] == tag1);

        if (__all(uniq)) {
            // no duplicates anywhere: every slot is its own first occurrence
            w[i0] = wv0 - lg;  dt[i0] = j;
            w[i1] = wv1 - lg;  dt[i1] = j;
        } else {
            // slow exact path: all-to-all shuffle dedup (count + first position)
            int cnt0 = 0, cnt1 = 0, first0 = 1 << 30, first1 = 1 << 30;
            #pragma unroll
            for (int t = 0; t < 32; ++t) {
                const int a0 = __shfl(i0, t);   // index at position 2t
                const int a1 = __shfl(i1, t);   // index at position 2t+1
                if (a0 == i0) { cnt0++; first0 = min(first0, 2 * t);     }
                if (a1 == i0) { cnt0++; first0 = min(first0, 2 * t + 1); }
                if (a0 == i1) { cnt1++; first1 = min(first1, 2 * t);     }
                if (a1 == i1) { cnt1++; first1 = min(first1, 2 * t + 1); }
            }
            // first occurrence writes decayed value minus lr*g per occurrence
            if (first0 == 2 * lane)     { w[i0] = wv0 - lg * (float)cnt0; dt[i0] = j; }
            if (first1 == 2 * lane + 1) { w[i1] = wv1 - lg * (float)cnt1; dt[i1] = j; }
        }

        b -= lg;

        // ensure this sample's scatter is ordered before next sample's gather
        asm volatile("s_wait_storecnt 0" ::: "memory");

        // hide next sample's index fetch under this iteration's tail
        if (j + 1 < B) {
            __builtin_prefetch(&idx[(j + 1) * 64 + 2 * lane], 0, 0);
        }
    }
}

extern "C" void kernel_launch(void* const* d_in, const int* in_sizes, int n_in,
                              void* d_out, int out_size, void* d_ws, size_t ws_size,
                              hipStream_t stream)
{
    // setup_inputs order: X_w_indices, X_d, y, w, b, decay_times
    const int*   X_w   = (const int*)  d_in[0];
    const float* y     = (const float*)d_in[2];
    const float* w_in  = (const float*)d_in[3];
    const float* b_in  = (const float*)d_in[4];
    const int*   dt_in = (const int*)  d_in[5];
    float*       out   = (float*)d_out;

    const int B  = in_sizes[2];   // 1024 samples
    const int Dn = in_sizes[3];   // 16M weights

    // state lives in workspace: never mutate inputs (harness replays without restore)
    float* w_ws  = (float*)d_ws;
    int*   dt_ws = (int*)((char*)d_ws + (size_t)Dn * sizeof(float));

    hipMemcpyAsync(w_ws,  w_in,  (size_t)Dn * sizeof(float),
                   hipMemcpyDeviceToDevice, stream);
    hipMemcpyAsync(dt_ws, dt_in, (size_t)Dn * sizeof(int),
                   hipMemcpyDeviceToDevice, stream);

    wide_sgd_seq_kernel<<<1, 32, 0, stream>>>(X_w, y, b_in, w_ws, dt_ws, out, B);
}